// CausalSelfAttention2d_83958020703018
// MI455X (gfx1250) — compile-verified
//
#include <hip/hip_runtime.h>
#include <math.h>

// CDNA5 (gfx1250) fp32 WMMA path: V_WMMA_F32_16X16X4_F32
// A: 16x4 f32 (2 VGPR/lane), B: 4x16 f32 (2 VGPR/lane), C/D: 16x16 f32 (8 VGPR/lane)
typedef __attribute__((ext_vector_type(2))) float v2f;
typedef __attribute__((ext_vector_type(8))) float v8f;

#define B_   8
#define CIN  256
#define S_   1024
#define KD   512
#define OC   512
#define NH   8
#define HD   64

__device__ __forceinline__ v8f wmma_f32(v2f a, v2f b, v8f c) {
  // 8 args: (neg_a, A, neg_b, B, c_mod, C, reuse_a, reuse_b)
  return __builtin_amdgcn_wmma_f32_16x16x4_f32(
      false, a, false, b, (short)0, c, false, false);
}

// ---------------------------------------------------------------------------
// Kernel 1: fused QKV projection.  x[b][s][c] = input[b][c][s] (implicit
// transpose folded into the A-fragment load).  Each wave computes a 16x64
// output strip; grid.x selects {Q,K,V} x column tile.
// A-frag layout (16x4 f32): lane L holds M=L%16, VGPR j holds K=2*(L/16)+j.
// B-frag layout (4x16 f32): lane L holds N=L%16, VGPR j holds K=2*(L/16)+j.
// ---------------------------------------------------------------------------
__global__ __launch_bounds__(256) void qkv_kernel(
    const float* __restrict__ input,
    const float* __restrict__ qw, const float* __restrict__ qb,
    const float* __restrict__ kw, const float* __restrict__ kb,
    const float* __restrict__ vw, const float* __restrict__ vb,
    float* __restrict__ Q, float* __restrict__ K, float* __restrict__ V)
{
  const int lane = threadIdx.x & 31;
  const int wave = threadIdx.x >> 5;
  const int hi16 = lane >> 4;      // 0 or 1 (which 16-lane half)
  const int l16  = lane & 15;

  const int ct    = blockIdx.x;            // 0..23
  const int which = ct >> 3;               // 0=Q, 1=K, 2=V
  const int col0  = (ct & 7) * 64;         // column base within 512
  const int row0  = blockIdx.y * 128 + wave * 16;
  const int b     = blockIdx.z;

  const float* w    = (which == 0) ? qw : (which == 1) ? kw : vw;
  const float* bias = (which == 0) ? qb : (which == 1) ? kb : vb;
  float*       out  = (which == 0) ? Q  : (which == 1) ? K  : V;

  const float* xb = input + (size_t)b * CIN * S_;   // input[b][c][s]
  const int m = row0 + l16;                          // A row handled by lane

  v8f acc0 = {}, acc1 = {}, acc2 = {}, acc3 = {};

  for (int k0 = 0; k0 < CIN; k0 += 4) {
    const int ka = k0 + 2 * hi16;
    v2f a;
    a.x = xb[(size_t)ka       * S_ + m];   // A[m][2*half]
    a.y = xb[(size_t)(ka + 1) * S_ + m];   // A[m][2*half+1]

    const float* wp = w + (size_t)(col0 + l16) * CIN + ka;  // contiguous in C
    v2f b0 = *(const v2f*)(wp + 0 * 16 * CIN);
    v2f b1 = *(const v2f*)(wp + 1 * 16 * CIN);
    v2f b2 = *(const v2f*)(wp + 2 * 16 * CIN);
    v2f b3 = *(const v2f*)(wp + 3 * 16 * CIN);

    acc0 = wmma_f32(a, b0, acc0);
    acc1 = wmma_f32(a, b1, acc1);
    acc2 = wmma_f32(a, b2, acc2);
    acc3 = wmma_f32(a, b3, acc3);
  }

  // Bias + store.  C/D layout: lane L, VGPR r -> row = r + 8*(L/16), col = L%16.
  float* ob = out + (size_t)b * S_ * KD;
  #pragma unroll
  for (int t = 0; t < 4; ++t) {
    v8f acc = (t == 0) ? acc0 : (t == 1) ? acc1 : (t == 2) ? acc2 : acc3;
    const int n  = col0 + t * 16 + l16;
    const float bv = bias[n];
    #pragma unroll
    for (int r = 0; r < 8; ++r) {
      const int s = row0 + r + 8 * hi16;
      ob[(size_t)s * KD + n] = acc[r] + bv;
    }
  }
}

// ---------------------------------------------------------------------------
// Kernel 2: causal flash attention per (b, head).  Each wave independently
// owns a 16-query strip: online softmax over causal key blocks of 16.
// P (probability tile) is round-tripped through a private LDS tile to convert
// the WMMA C/D layout into the A layout for the P@V product.
// ---------------------------------------------------------------------------
__global__ __launch_bounds__(256) void attn_kernel(
    const float* __restrict__ Q, const float* __restrict__ K,
    const float* __restrict__ V, float* __restrict__ out)
{
  __shared__ float plds[8][16 * 16];       // one 16x16 tile per wave (8 KB)

  const int lane = threadIdx.x & 31;
  const int wave = threadIdx.x >> 5;
  const int hi16 = lane >> 4;
  const int l16  = lane & 15;

  const int i0 = (blockIdx.x * 8 + wave) * 16;   // query strip base
  const int h  = blockIdx.y;
  const int b  = blockIdx.z;

  const float* Qb = Q + (size_t)b * S_ * KD + (size_t)h * HD;
  const float* Kb = K + (size_t)b * S_ * KD + (size_t)h * HD;
  const float* Vb = V + (size_t)b * S_ * KD + (size_t)h * HD;

  // Preload Q fragments: A[m][d], d = 4t + 2*half + {0,1}
  v2f qa[16];
  {
    const float* qrow = Qb + (size_t)(i0 + l16) * KD + 2 * hi16;
    #pragma unroll
    for (int t = 0; t < 16; ++t) qa[t] = *(const v2f*)(qrow + 4 * t);
  }

  v8f o0 = {}, o1 = {}, o2 = {}, o3 = {};
  float mrow[8], lrow[8];
  #pragma unroll
  for (int r = 0; r < 8; ++r) { mrow[r] = -INFINITY; lrow[r] = 0.f; }

  float* pl = &plds[wave][0];

  for (int j0 = 0; j0 <= i0; j0 += 16) {
    // ---- scores = (Q . K^T) for this 16x16 tile, K-dim = 64 -> 16 WMMAs ----
    v8f sc = {};
    const float* krow = Kb + (size_t)(j0 + l16) * KD + 2 * hi16;
    #pragma unroll
    for (int t = 0; t < 16; ++t) {
      v2f kb2 = *(const v2f*)(krow + 4 * t);   // B[d][n] = K[j0+n][d]
      sc = wmma_f32(qa[t], kb2, sc);
    }

    // ---- scale + causal mask (diagonal block only) ----
    float p[8];
    #pragma unroll
    for (int r = 0; r < 8; ++r) {
      float v = sc[r] * 0.125f;                      // 1/sqrt(64)
      if (j0 == i0 && (j0 + l16) > (i0 + r + 8 * hi16)) v = -INFINITY;
      p[r] = v;
    }

    // ---- online softmax (row stats across the 16-lane half) ----
    float scale[8];
    #pragma unroll
    for (int r = 0; r < 8; ++r) {
      float rm = p[r];
      rm = fmaxf(rm, __shfl_xor(rm, 1, 32));
      rm = fmaxf(rm, __shfl_xor(rm, 2, 32));
      rm = fmaxf(rm, __shfl_xor(rm, 4, 32));
      rm = fmaxf(rm, __shfl_xor(rm, 8, 32));
      const float mn = fmaxf(mrow[r], rm);
      const float sc_ = __expf(mrow[r] - mn);
      const float pv  = __expf(p[r] - mn);
      p[r] = pv;
      float rs = pv;
      rs += __shfl_xor(rs, 1, 32);
      rs += __shfl_xor(rs, 2, 32);
      rs += __shfl_xor(rs, 4, 32);
      rs += __shfl_xor(rs, 8, 32);
      lrow[r] = lrow[r] * sc_ + rs;
      mrow[r] = mn;
      scale[r] = sc_;
    }
    #pragma unroll
    for (int r = 0; r < 8; ++r) {
      o0[r] *= scale[r]; o1[r] *= scale[r];
      o2[r] *= scale[r]; o3[r] *= scale[r];
    }

    // ---- stage P through LDS: C/D layout -> row-major 16x16 ----
    #pragma unroll
    for (int r = 0; r < 8; ++r)
      pl[(r + 8 * hi16) * 16 + l16] = p[r];
    asm volatile("s_wait_dscnt 0" ::: "memory");   // same-wave RAW on LDS

    // ---- out += P @ V_block  (M=16, N=64, K=16 -> 16 WMMAs) ----
    #pragma unroll
    for (int t2 = 0; t2 < 4; ++t2) {
      v2f pa = *(const v2f*)(pl + l16 * 16 + 4 * t2 + 2 * hi16); // A[m][k]
      const int kk = j0 + 4 * t2 + 2 * hi16;
      const float* vrow = Vb + (size_t)kk * KD;
      #pragma unroll
      for (int nt = 0; nt < 4; ++nt) {
        v2f vb2;
        vb2.x = vrow[      nt * 16 + l16];   // V[kk  ][n]
        vb2.y = vrow[KD +  nt * 16 + l16];   // V[kk+1][n]
        v8f& oo = (nt == 0) ? o0 : (nt == 1) ? o1 : (nt == 2) ? o2 : o3;
        oo = wmma_f32(pa, vb2, oo);
      }
    }
  }

  // ---- epilogue: normalize and scatter to (b, out_ch, h, w) layout ----
  float* ob = out + ((size_t)b * OC + (size_t)h * HD) * S_;
  #pragma unroll
  for (int t = 0; t < 4; ++t) {
    v8f oo = (t == 0) ? o0 : (t == 1) ? o1 : (t == 2) ? o2 : o3;
    const int d = t * 16 + l16;
    #pragma unroll
    for (int r = 0; r < 8; ++r) {
      const int s = i0 + r + 8 * hi16;
      ob[(size_t)d * S_ + s] = oo[r] / lrow[r];
    }
  }
}

extern "C" void kernel_launch(void* const* d_in, const int* in_sizes, int n_in,
                              void* d_out, int out_size, void* d_ws, size_t ws_size,
                              hipStream_t stream) {
  (void)in_sizes; (void)n_in; (void)out_size; (void)ws_size;
  const float* input = (const float*)d_in[0];
  const float* qw = (const float*)d_in[1];
  const float* qb = (const float*)d_in[2];
  const float* kw = (const float*)d_in[3];
  const float* kb = (const float*)d_in[4];
  const float* vw = (const float*)d_in[5];
  const float* vb = (const float*)d_in[6];
  float* out = (float*)d_out;

  // Workspace: Q, K, V each (B, S, 512) fp32 = 16 MB -> 48 MB total.
  float* Q = (float*)d_ws;
  float* K = Q + (size_t)B_ * S_ * KD;
  float* V = K + (size_t)B_ * S_ * KD;

  dim3 g1(24, 8, 8);   // {Q|K|V} x col-tile, row-tile, batch
  qkv_kernel<<<g1, 256, 0, stream>>>(input, qw, qb, kw, kb, vw, vb, Q, K, V);

  dim3 g2(8, 8, 8);    // query-strip group, head, batch
  attn_kernel<<<g2, 256, 0, stream>>>(Q, K, V, out);
}